// ClassifierRNN_22737556865439
// MI455X (gfx1250) — compile-verified
//
#include <hip/hip_runtime.h>
#include <stdint.h>

// Problem constants (from reference)
#define Bn 64
#define Tn 2048
#define En 300
#define H1n 64
#define H2n 32
#define Mrows (Bn * Tn) // 131072 tokens

typedef __attribute__((ext_vector_type(2))) float v2f;
typedef __attribute__((ext_vector_type(8))) float v8f;
typedef __attribute__((ext_vector_type(4))) unsigned int u32x4;
typedef __attribute__((ext_vector_type(4))) int i32x4;
typedef __attribute__((ext_vector_type(8))) int i32x8;

// f32 WMMA: D(16x16,f32) = A(16x4,f32) x B(4x16,f32) + C  — full f32 fidelity
__device__ __forceinline__ v8f wmma4(v2f a, v2f b, v8f c) {
  return __builtin_amdgcn_wmma_f32_16x16x4_f32(false, a, false, b, (short)0, c,
                                               false, false);
}

// Fast activations on the serialized scan critical path: use CDNA5 hardware
// transcendentals (v_exp_f32 / v_rcp_f32 / v_tanh_f32) instead of OCML
// polynomials + IEEE divides.
__device__ __forceinline__ float sigmoidf_(float x) {
  return __builtin_amdgcn_rcpf(1.0f +
                               __builtin_amdgcn_exp2f(-1.44269504f * x));
}
__device__ __forceinline__ float tanhf_(float x) {
#if __has_builtin(__builtin_amdgcn_tanhf)
  return __builtin_amdgcn_tanhf(x);
#else
  const float e = __builtin_amdgcn_exp2f(-2.88539008f * fabsf(x));
  const float r = (1.0f - e) * __builtin_amdgcn_rcpf(1.0f + e);
  return copysignf(r, x);
#endif
}

// TDM: 2D tile (tile1 rows x tile0 f32 elems, row stride stride0 elems) from
// global -> LDS, with LDS row padding (pad_amount_code=1 -> +2 dwords/row)
// to make the strided WMMA C-layout LDS accesses bank-conflict free.
__device__ __forceinline__ void tdm_load_2d(uint32_t lds_byte, const void* gptr,
                                            uint32_t tile0, uint32_t tile1,
                                            uint64_t stride0,
                                            uint32_t pad_interval_code,
                                            uint32_t pad_amount_code) {
  uint64_t ga = (uint64_t)(uintptr_t)gptr;
  u32x4 g0;
  g0[0] = 1u;                                             // count=1, user D#
  g0[1] = lds_byte;                                       // lds_addr
  g0[2] = (uint32_t)ga;                                   // global addr lo
  g0[3] = ((uint32_t)(ga >> 32) & 0x01FFFFFFu) | (2u << 30); // hi | type=2
  i32x8 g1;
  // data_size=2 (4B), pad_enable, pad_interval, pad_amount; wg_mask=0 (no cluster)
  g1[0] = (int)((2u << 16) | (1u << 20) | (pad_interval_code << 22) |
                (pad_amount_code << 25));
  g1[1] = (int)((tile0 & 0xFFFFu) << 16);                 // tensor_dim0 lo16
  g1[2] = (int)((tile0 >> 16) | ((tile1 & 0xFFFFu) << 16)); // td0 hi | td1 lo
  g1[3] = (int)((tile1 >> 16) | (tile0 << 16));           // td1 hi | tile_dim0
  g1[4] = (int)tile1;                                     // tile_dim1, tile_dim2=0
  g1[5] = (int)(uint32_t)stride0;                         // dim0_stride lo32
  g1[6] = (int)((uint32_t)(stride0 >> 32) & 0xFFFFu);     // dim0_stride hi16
  g1[7] = 0;
  i32x4 z4 = {0, 0, 0, 0};
#if __clang_major__ >= 23
  i32x8 z8 = {0, 0, 0, 0, 0, 0, 0, 0};
  __builtin_amdgcn_tensor_load_to_lds(g0, g1, z4, z4, z8, 0);
#else
  __builtin_amdgcn_tensor_load_to_lds(g0, g1, z4, z4, 0);
#endif
}

// ---------------------------------------------------------------------------
// Kernel 1: fused embedding gather + layer-1 input projection (both dirs)
// G1[m][n] = emb[ids[m]][:] . Wcat[n][:] + (bih+bhh)[n],  n<256 fwd, n>=256 bwd
// ---------------------------------------------------------------------------
__global__ __launch_bounds__(256) void k_gemm1(
    const int* __restrict__ ids, const float* __restrict__ emb,
    const float* __restrict__ wf, const float* __restrict__ wb,
    const float* __restrict__ bihf, const float* __restrict__ bhhf,
    const float* __restrict__ bihb, const float* __restrict__ bhhb,
    float* __restrict__ g1out) {
  __shared__ float A[16 * En]; // 16 token rows staged (stride 300: conflict-free)
  const int tid = threadIdx.x, lane = tid & 31, wv = tid >> 5;
  const size_t m0 = (size_t)blockIdx.x * 16;
  for (int rr = 0; rr < 2; ++rr) {
    const int row = wv * 2 + rr;
    const int id = ids[m0 + row];
    const float* er = emb + (size_t)id * En;
    for (int c = lane; c < En; c += 32) A[row * En + c] = er[c];
  }
  __syncthreads();
  const int cl = lane & 15;
  const int khalf = (lane >> 4) << 1; // K offset 0 / 2 per half-wave
  const int rb = (lane >> 4) << 3;    // C-row base 0 / 8 per half-wave
  for (int nt = wv; nt < 32; nt += 8) {
    const int n = nt * 16 + cl;
    const float* wrow;
    float bias;
    if (n < 256) { wrow = wf + (size_t)n * En; bias = bihf[n] + bhhf[n]; }
    else { wrow = wb + (size_t)(n - 256) * En; bias = bihb[n - 256] + bhhb[n - 256]; }
    v8f acc;
    for (int r = 0; r < 8; ++r) acc[r] = bias;
    for (int k = 0; k < En; k += 4) { // 300 = 75 * 4
      v2f a = *(const v2f*)&A[cl * En + k + khalf];
      v2f b = *(const v2f*)&wrow[k + khalf];
      acc = wmma4(a, b, acc);
    }
    for (int r = 0; r < 8; ++r) g1out[(m0 + rb + r) * 512 + n] = acc[r];
  }
}

// ---------------------------------------------------------------------------
// Kernel 2: layer-1 recurrent scan, one block per direction (grid=2, 512 thr).
// Per step: gates[64x256] = Gin(TDM-streamed) + H[64x64] @ Whh^T, then LSTM cell.
// Whh^T fragments pinned in VGPRs; H in LDS (stride 68, conflict-free A reads);
// cell state in registers; gate slab double-buffered via TDM + s_wait_tensorcnt;
// per-step pad masks staged into LDS by wave 1 (overlapped with WMMA phase).
// ---------------------------------------------------------------------------
__global__ __launch_bounds__(512) void k_scan1(
    const int* __restrict__ ids, const float* __restrict__ whhf,
    const float* __restrict__ whhb, const float* __restrict__ g1,
    float* __restrict__ o1) {
  const int dir = blockIdx.x; // 0 fwd, 1 bwd
  const float* whh = dir ? whhb : whhf;
  const int gate_off = dir ? 256 : 0;
  const int out_off = dir ? H1n : 0;

  __shared__ float H[64 * 68];
  __shared__ float Gin[2][64 * 258]; // 256 gates + 2-dword TDM pad per row
  __shared__ int Msk[64];            // per-step pad masks

  const int tid = threadIdx.x, lane = tid & 31, wv = tid >> 5;
  for (int i = tid; i < 64 * 68; i += 512) H[i] = 0.0f;

  const int cl = lane & 15;
  const int khalf = (lane >> 4) << 1;
  const int nb = wv * 16; // this wave's 16 gate columns
  v2f Bf[16];             // Whh^T fragments, K=0..63
#pragma unroll
  for (int j = 0; j < 16; ++j)
    Bf[j] = *(const v2f*)&whh[(nb + cl) * 64 + 4 * j + khalf];

  float creg[8];
#pragma unroll
  for (int p = 0; p < 8; ++p) creg[p] = 0.0f;
  __syncthreads();

  const int tfirst = dir ? (Tn - 1) : 0;
  if (wv == 0)
    tdm_load_2d((uint32_t)(uintptr_t)&Gin[0][0],
                g1 + (size_t)tfirst * 512 + gate_off, 256, 64,
                (uint64_t)Tn * 512, 7, 1);

  for (int s = 0; s < Tn; ++s) {
    const int t = dir ? (Tn - 1 - s) : s;
    const int cur = s & 1;
    if (wv == 0) __builtin_amdgcn_s_wait_tensorcnt(0); // buf[cur] landed
    __syncthreads(); // data visible + prev-step consumers of buf[cur^1] done
    if (wv == 0 && s + 1 < Tn) {
      const int tnext = dir ? (t - 1) : (t + 1);
      tdm_load_2d((uint32_t)(uintptr_t)&Gin[cur ^ 1][0],
                  g1 + (size_t)tnext * 512 + gate_off, 256, 64,
                  (uint64_t)Tn * 512, 7, 1);
    }
    if (wv == 1) { // stage this step's masks (visible after mid barrier)
      Msk[lane] = ids[lane * Tn + t];
      Msk[lane + 32] = ids[(lane + 32) * Tn + t];
    }
    float* gin = &Gin[cur][0];
#pragma unroll
    for (int mt = 0; mt < 4; ++mt) {
      const int rb = mt * 16 + ((lane >> 4) << 3);
      v8f acc;
#pragma unroll
      for (int r = 0; r < 8; ++r) acc[r] = gin[(rb + r) * 258 + nb + cl];
#pragma unroll
      for (int j = 0; j < 16; ++j) {
        v2f a = *(const v2f*)&H[(mt * 16 + cl) * 68 + 4 * j + khalf];
        acc = wmma4(a, Bf[j], acc);
      }
#pragma unroll
      for (int r = 0; r < 8; ++r) gin[(rb + r) * 258 + nb + cl] = acc[r];
    }
    __syncthreads(); // all gate sums + masks in LDS
#pragma unroll
    for (int p = 0; p < 8; ++p) {
      const int idx = p * 512 + tid;
      const int b = idx >> 6, j = idx & 63;
      const bool m = Msk[b] != 0;
      const float* gr = gin + b * 258;
      const float i_ = sigmoidf_(gr[j]);
      const float f_ = sigmoidf_(gr[64 + j]);
      const float g_ = tanhf_(gr[128 + j]);
      const float o_ = sigmoidf_(gr[192 + j]);
      const float cn = f_ * creg[p] + i_ * g_;
      const float hn = o_ * tanhf_(cn);
      if (m) { creg[p] = cn; H[b * 68 + j] = hn; }
      o1[((size_t)b * Tn + t) * 128 + out_off + j] = m ? hn : 0.0f;
    } // next-iteration top barrier orders H writes vs WMMA reads
  }
}

// ---------------------------------------------------------------------------
// Kernel 3: layer-2 input projection (both dirs): G2 = o1 @ [Wih2f;Wih2b]^T + b
// ---------------------------------------------------------------------------
__global__ __launch_bounds__(256) void k_gemm2(
    const float* __restrict__ o1, const float* __restrict__ wf,
    const float* __restrict__ wb, const float* __restrict__ bihf,
    const float* __restrict__ bhhf, const float* __restrict__ bihb,
    const float* __restrict__ bhhb, float* __restrict__ g2out) {
  __shared__ float A[16 * 132];
  const int tid = threadIdx.x, lane = tid & 31, wv = tid >> 5;
  const size_t m0 = (size_t)blockIdx.x * 16;
  for (int rr = 0; rr < 2; ++rr) {
    const int row = wv * 2 + rr;
    const float* orow = o1 + (m0 + row) * 128;
    for (int c = lane; c < 128; c += 32) A[row * 132 + c] = orow[c];
  }
  __syncthreads();
  const int cl = lane & 15;
  const int khalf = (lane >> 4) << 1;
  const int rb = (lane >> 4) << 3;
  for (int nt = wv; nt < 16; nt += 8) {
    const int n = nt * 16 + cl;
    const float* wrow;
    float bias;
    if (n < 128) { wrow = wf + (size_t)n * 128; bias = bihf[n] + bhhf[n]; }
    else { wrow = wb + (size_t)(n - 128) * 128; bias = bihb[n - 128] + bhhb[n - 128]; }
    v8f acc;
    for (int r = 0; r < 8; ++r) acc[r] = bias;
    for (int k = 0; k < 128; k += 4) {
      v2f a = *(const v2f*)&A[cl * 132 + k + khalf];
      v2f b = *(const v2f*)&wrow[k + khalf];
      acc = wmma4(a, b, acc);
    }
    for (int r = 0; r < 8; ++r) g2out[(m0 + rb + r) * 256 + n] = acc[r];
  }
}

// ---------------------------------------------------------------------------
// Kernel 4: layer-2 scan. Block 0: full forward scan (WMMA+TDM like layer 1),
// emits h at t=T-1 only. Block 1: backward dir contributes o2[:,-1,:] at its
// FIRST reverse step (zero state) -> single-step closed form, no scan needed.
// ---------------------------------------------------------------------------
__global__ __launch_bounds__(256) void k_scan2(
    const int* __restrict__ ids, const float* __restrict__ whh2f,
    const float* __restrict__ g2, float* __restrict__ last) {
  const int tid = threadIdx.x;
  if (blockIdx.x == 1) {
    for (int idx = tid; idx < Bn * H2n; idx += 256) {
      const int b = idx >> 5, j = idx & 31;
      const size_t base = ((size_t)b * Tn + (Tn - 1)) * 256 + 128;
      const float i_ = sigmoidf_(g2[base + j]);
      const float g_ = tanhf_(g2[base + 64 + j]);
      const float o_ = sigmoidf_(g2[base + 96 + j]);
      const float hn = o_ * tanhf_(i_ * g_); // f*c0 = 0
      const bool m = ids[b * Tn + (Tn - 1)] != 0;
      last[b * 64 + 32 + j] = m ? hn : 0.0f;
    }
    return;
  }
  __shared__ float H[64 * 36];
  __shared__ float Gin[2][64 * 130]; // 128 gates + 2-dword TDM pad
  __shared__ int Msk[64];
  const int lane = tid & 31, wv = tid >> 5;
  for (int i = tid; i < 64 * 36; i += 256) H[i] = 0.0f;
  const int cl = lane & 15;
  const int khalf = (lane >> 4) << 1;
  const int nb = wv * 16;
  v2f Bf[8];
#pragma unroll
  for (int j = 0; j < 8; ++j)
    Bf[j] = *(const v2f*)&whh2f[(nb + cl) * 32 + 4 * j + khalf];
  float creg[8];
#pragma unroll
  for (int p = 0; p < 8; ++p) creg[p] = 0.0f;
  __syncthreads();
  if (wv == 0)
    tdm_load_2d((uint32_t)(uintptr_t)&Gin[0][0], g2, 128, 64,
                (uint64_t)Tn * 256, 6, 1);
  for (int s = 0; s < Tn; ++s) {
    const int cur = s & 1;
    if (wv == 0) __builtin_amdgcn_s_wait_tensorcnt(0);
    __syncthreads();
    if (wv == 0 && s + 1 < Tn)
      tdm_load_2d((uint32_t)(uintptr_t)&Gin[cur ^ 1][0],
                  g2 + (size_t)(s + 1) * 256, 128, 64, (uint64_t)Tn * 256, 6, 1);
    if (wv == 1) {
      Msk[lane] = ids[lane * Tn + s];
      Msk[lane + 32] = ids[(lane + 32) * Tn + s];
    }
    float* gin = &Gin[cur][0];
#pragma unroll
    for (int mt = 0; mt < 4; ++mt) {
      const int rb = mt * 16 + ((lane >> 4) << 3);
      v8f acc;
#pragma unroll
      for (int r = 0; r < 8; ++r) acc[r] = gin[(rb + r) * 130 + nb + cl];
#pragma unroll
      for (int j = 0; j < 8; ++j) {
        v2f a = *(const v2f*)&H[(mt * 16 + cl) * 36 + 4 * j + khalf];
        acc = wmma4(a, Bf[j], acc);
      }
#pragma unroll
      for (int r = 0; r < 8; ++r) gin[(rb + r) * 130 + nb + cl] = acc[r];
    }
    __syncthreads();
#pragma unroll
    for (int p = 0; p < 8; ++p) {
      const int idx = p * 256 + tid;
      const int b = idx >> 5, j = idx & 31;
      const bool m = Msk[b] != 0;
      const float* gr = gin + b * 130;
      const float i_ = sigmoidf_(gr[j]);
      const float f_ = sigmoidf_(gr[32 + j]);
      const float g_ = tanhf_(gr[64 + j]);
      const float o_ = sigmoidf_(gr[96 + j]);
      const float cn = f_ * creg[p] + i_ * g_;
      const float hn = o_ * tanhf_(cn);
      if (m) { creg[p] = cn; H[b * 36 + j] = hn; }
      if (s == Tn - 1) last[b * 64 + j] = m ? hn : 0.0f;
    }
  }
}

// ---------------------------------------------------------------------------
// Kernel 5: MLP head: out = relu(last @ w1^T + b1) @ w2^T + b2   (tiny)
// ---------------------------------------------------------------------------
__global__ __launch_bounds__(64) void k_head(
    const float* __restrict__ last, const float* __restrict__ w1,
    const float* __restrict__ b1, const float* __restrict__ w2,
    const float* __restrict__ b2, float* __restrict__ out) {
  const int b = threadIdx.x;
  float x[64];
  for (int i = 0; i < 64; ++i) x[i] = last[b * 64 + i];
  float h[20];
  for (int l = 0; l < 20; ++l) {
    float s = b1[l];
    for (int i = 0; i < 64; ++i) s += x[i] * w1[l * 64 + i];
    h[l] = s > 0.0f ? s : 0.0f;
  }
  for (int k = 0; k < 4; ++k) {
    float s = b2[k];
    for (int l = 0; l < 20; ++l) s += h[l] * w2[k * 20 + l];
    out[b * 4 + k] = s;
  }
}

extern "C" void kernel_launch(void* const* d_in, const int* in_sizes, int n_in,
                              void* d_out, int out_size, void* d_ws,
                              size_t ws_size, hipStream_t stream) {
  (void)in_sizes; (void)n_in; (void)out_size; (void)ws_size;
  const int* ids = (const int*)d_in[0];
  const float* emb = (const float*)d_in[1];
  const float* w_ih1f = (const float*)d_in[2];
  const float* w_hh1f = (const float*)d_in[3];
  const float* b_ih1f = (const float*)d_in[4];
  const float* b_hh1f = (const float*)d_in[5];
  const float* w_ih1b = (const float*)d_in[6];
  const float* w_hh1b = (const float*)d_in[7];
  const float* b_ih1b = (const float*)d_in[8];
  const float* b_hh1b = (const float*)d_in[9];
  const float* w_ih2f = (const float*)d_in[10];
  const float* w_hh2f = (const float*)d_in[11];
  const float* b_ih2f = (const float*)d_in[12];
  const float* b_hh2f = (const float*)d_in[13];
  const float* w_ih2b = (const float*)d_in[14];
  const float* w_hh2b = (const float*)d_in[15]; // unused: bwd L2 single-step f*c0=0
  const float* b_ih2b = (const float*)d_in[16];
  const float* b_hh2b = (const float*)d_in[17];
  const float* w1 = (const float*)d_in[18];
  const float* b1 = (const float*)d_in[19];
  const float* w2 = (const float*)d_in[20];
  const float* b2 = (const float*)d_in[21];
  (void)w_hh2b;

  // Workspace layout (floats): G1 [M][512] | o1 [M][128] | last [64][64]
  // G2 [M][256] reuses G1's memory after the layer-1 scans complete.
  float* wsf = (float*)d_ws;
  float* g1 = wsf;
  float* o1 = wsf + (size_t)Mrows * 512;
  float* g2 = wsf;
  float* last = wsf + (size_t)Mrows * 640;

  k_gemm1<<<Mrows / 16, 256, 0, stream>>>(ids, emb, w_ih1f, w_ih1b, b_ih1f,
                                          b_hh1f, b_ih1b, b_hh1b, g1);
  k_scan1<<<2, 512, 0, stream>>>(ids, w_hh1f, w_hh1b, g1, o1);
  k_gemm2<<<Mrows / 16, 256, 0, stream>>>(o1, w_ih2f, w_ih2b, b_ih2f, b_hh2f,
                                          b_ih2b, b_hh2b, g2);
  k_scan2<<<2, 256, 0, stream>>>(ids, w_hh2f, g2, last);
  k_head<<<1, 64, 0, stream>>>(last, w1, b1, w2, b2, (float*)d_out);
}